// HarmonicGenerator_23794118820311
// MI455X (gfx1250) — compile-verified
//
#include <hip/hip_runtime.h>
#include <hip/hip_bf16.h>
#include <math.h>

// ---------------- problem constants (from reference) ----------------
constexpr int BB    = 4;
constexpr int TT    = 800;
constexpr int NH    = 80;
constexpr int HOP   = 240;
constexpr int NFFT  = 1024;
constexpr int WIN   = 1024;
constexpr float SRf = 24000.0f;
constexpr int TOTAL = (TT - 1) * HOP + WIN;       // 192784 samples per batch
constexpr int PAD   = NFFT / 2;                   // 512
constexpr int NBINS = NFFT / 2 + 1;               // 513
constexpr int MGEMM = BB * TT;                    // 3200 rows
constexpr int NOUT  = NBINS * 2;                  // 1026 (re,im interleaved cols)
constexpr int NPAD  = 1040;                       // 65 * 16 column padding
constexpr int NTILE = NPAD / 16;                  // 65
constexpr int CHUNK = 4096;                       // scan chunk (256 thr * 16)
constexpr int NCH   = (TOTAL + CHUNK - 1) / CHUNK; // 48
constexpr int ASTRIDE = 1028;                     // LDS row stride (floats): 1028%64=4
                                                  // -> 16 lanes hit distinct banks

constexpr float PIF = 3.14159265358979323846f;

// ---------------- workspace layout ----------------
constexpr size_t alignup(size_t x) { return (x + 255) & ~(size_t)255; }
constexpr size_t OFF_AUDIO = 0;
constexpr size_t OFF_CSUM  = alignup(OFF_AUDIO + (size_t)BB * TOTAL * 4);
constexpr size_t OFF_CBASE = alignup(OFF_CSUM  + (size_t)BB * NCH * 4);
constexpr size_t OFF_MAX   = alignup(OFF_CBASE + (size_t)BB * NCH * 4);
constexpr size_t OFF_AMAT  = alignup(OFF_MAX   + (size_t)BB * 4);
constexpr size_t OFF_BMAT  = alignup(OFF_AMAT  + (size_t)MGEMM * NFFT * 4);
// total ws need: OFF_BMAT + 1024*1040*4  ~= 20.5 MB

typedef __attribute__((ext_vector_type(2))) float v2f;
typedef __attribute__((ext_vector_type(8))) float v8f;

// phase increment d_phi(b, s) = 2*pi*f0_interp/SR
__device__ __forceinline__ float dphi_at(const float* __restrict__ f0, int b, int s) {
    int prev = s / HOP;
    if (prev > TT - 2) prev = TT - 2;
    float w = (float)(s - prev * HOP) * (1.0f / (float)HOP);
    w = fminf(w, 1.0f);
    float fp = f0[b * TT + prev];
    float fn = f0[b * TT + prev + 1];
    float f  = fmaf(w, fn - fp, fp);
    f = fmaxf(f, 0.0f);
    return f * (2.0f * PIF / SRf);
}

// ---------------- kernel 1: per-chunk phase sums ----------------
__global__ void k_chunk_sums(const float* __restrict__ f0, float* __restrict__ csum) {
    int c = blockIdx.x, b = blockIdx.y, t = threadIdx.x;
    int base = c * CHUNK + t * 16;
    float s = 0.0f;
    for (int i = 0; i < 16; ++i) {
        int ss = base + i;
        if (ss < TOTAL) s += dphi_at(f0, b, ss);
    }
    __shared__ float sm[256];
    sm[t] = s;
    __syncthreads();
    for (int off = 128; off > 0; off >>= 1) {
        if (t < off) sm[t] += sm[t + off];
        __syncthreads();
    }
    if (t == 0) csum[b * NCH + c] = sm[0];
}

// ---------------- kernel 2: exclusive scan of chunk sums (+init max) ---------
__global__ void k_scan(const float* __restrict__ csum, float* __restrict__ cbase,
                       unsigned* __restrict__ maxbuf) {
    int b = threadIdx.x;
    if (b < BB) {
        float run = 0.0f;
        for (int c = 0; c < NCH; ++c) {
            cbase[b * NCH + c] = run;
            run += csum[b * NCH + c];
        }
        maxbuf[b] = 0u;
    }
}

// ---------------- kernel 3: synthesis (local scan + harmonic sum) ------------
__global__ void k_synth(const float* __restrict__ f0, const float* __restrict__ amps,
                        const float* __restrict__ cbase, float* __restrict__ audio,
                        unsigned* __restrict__ maxbuf) {
    int c = blockIdx.x, b = blockIdx.y, t = threadIdx.x;
    int base = c * CHUNK + t * 16;

    float d[16];
    float run = 0.0f;
    for (int i = 0; i < 16; ++i) {
        int s = base + i;
        float v = (s < TOTAL) ? dphi_at(f0, b, s) : 0.0f;
        run += v;
        d[i] = run;                      // inclusive within thread
    }
    __shared__ float sm[256];
    sm[t] = run;
    __syncthreads();
    // Hillis-Steele inclusive scan of per-thread totals
    for (int off = 1; off < 256; off <<= 1) {
        float v = (t >= off) ? sm[t - off] : 0.0f;
        __syncthreads();
        if (t >= off) sm[t] += v;
        __syncthreads();
    }
    float prefix = cbase[b * NCH + c] + (sm[t] - run);  // exclusive for this thread
    __syncthreads();

    float maxv = 0.0f;
    for (int i = 0; i < 16; ++i) {
        int s = base + i;
        if (s >= TOTAL) break;
        float ph = prefix + d[i];
        int prev = s / HOP;
        if (prev > TT - 2) prev = TT - 2;
        float w = fminf((float)(s - prev * HOP) * (1.0f / (float)HOP), 1.0f);
        const float* ap = amps + ((size_t)b * TT + prev) * NH;
        const float* an = ap + NH;
        float sph, cph;
        sincosf(ph, &sph, &cph);
        float c2 = 2.0f * cph;
        // Chebyshev recurrence: sin((h+1)x) = 2cos(x) sin(hx) - sin((h-1)x)
        float sprev = 0.0f, scur = sph, acc = 0.0f;
        #pragma unroll 8
        for (int h = 0; h < NH; ++h) {
            float a = ap[h];
            a = fmaf(w, an[h] - a, a);
            acc = fmaf(a, scur, acc);
            float snext = fmaf(c2, scur, -sprev);
            sprev = scur;
            scur  = snext;
        }
        audio[(size_t)b * TOTAL + s] = acc;
        maxv = fmaxf(maxv, fabsf(acc));
    }

    __syncthreads();
    sm[t] = maxv;
    __syncthreads();
    for (int off = 128; off > 0; off >>= 1) {
        if (t < off) sm[t] = fmaxf(sm[t], sm[t + off]);
        __syncthreads();
    }
    if (t == 0) atomicMax(&maxbuf[b], __float_as_uint(sm[0]));
}

// ---------------- kernel 4: normalize + reflect-pad + Hann window → A --------
__global__ void k_frames(const float* __restrict__ audio, const unsigned* __restrict__ maxbuf,
                         float* __restrict__ Amat) {
    int t = blockIdx.x;   // frame index 0..799
    int b = blockIdx.y;
    float mx  = fmaxf(__uint_as_float(maxbuf[b]), 1e-7f);
    float inv = 1.0f / mx;
    const float* au = audio + (size_t)b * TOTAL;
    float* row = Amat + ((size_t)b * TT + t) * NFFT;
    for (int n = threadIdx.x; n < NFFT; n += 256) {
        int j = t * HOP + n - PAD;
        if (j < 0) j = -j;                        // reflect left
        if (j >= TOTAL) j = 2 * (TOTAL - 1) - j;  // reflect right
        float win = 0.5f - 0.5f * cosf((2.0f * PIF / (float)WIN) * (float)n);
        row[n] = au[j] * inv * win;
    }
}

// ---------------- kernel 5: DFT coefficient matrix (K=1024 x NPAD) -----------
__global__ void k_dft(float* __restrict__ Bmat) {
    int n = blockIdx.x;                 // 0..1023 (time sample within frame)
    float* row = Bmat + (size_t)n * NPAD;
    for (int col = threadIdx.x; col < NPAD; col += 256) {
        float v = 0.0f;
        if (col < NOUT) {
            int k  = col >> 1;
            int kn = (k * n) & (NFFT - 1);        // exact phase mod N
            float ang = (float)kn * (2.0f * PIF / (float)NFFT);
            float s, c;
            sincosf(ang, &s, &c);
            v = (col & 1) ? -s : c;               // e^{-i 2pi kn/N}
        }
        row[col] = v;
    }
}

// ---------------- kernel 6: WMMA f32 GEMM (frames @ DFT) ---------------------
// C(3200 x 1026) = A(3200 x 1024) * B(1024 x 1026), tile 16x16, K-step 4.
// One 16-row A tile per block, staged in LDS via async global->LDS copies;
// 4 waves per block cover 4 adjacent N tiles from the shared tile.
// Output scattered into transposed complex layout (B, 513, 800) interleaved.
__global__ void __launch_bounds__(128)
k_gemm(const float* __restrict__ Amat, const float* __restrict__ Bmat,
       float* __restrict__ out) {
    __shared__ float smA[16 * ASTRIDE];           // ~64 KB, padded rows

    const int tid   = threadIdx.x;
    const int lane  = tid & 31;
    const int wave  = tid >> 5;
    const int mTile = blockIdx.y;                 // 0..199
    const int nTile = blockIdx.x * 4 + wave;      // 0..67 (guard < 65)
    const int m0 = mTile * 16;
    const int half = lane >> 4;                   // 0: lanes 0-15, 1: lanes 16-31
    const int l15  = lane & 15;

    // ---- async copy: 16x1024 A tile (contiguous 64KB in row-major A) -> LDS
    // Generic pointer to LDS: low 32 bits are the LDS byte offset (aperture in
    // the high bits), which is what the async-LDS VDST operand wants.
    unsigned lds_base = (unsigned)(size_t)(void*)&smA[0];
    const float* gsrc = Amat + (size_t)m0 * NFFT;
    #pragma unroll 4
    for (int i = 0; i < 32; ++i) {
        int chunk = i * 128 + tid;                // b128 chunk id, coalesced
        int row   = chunk >> 8;                   // 256 chunks per 1024-f row
        int inrow = chunk & 255;
        unsigned laddr = lds_base + (unsigned)(row * ASTRIDE + inrow * 4) * 4u;
        const float* ga = gsrc + (size_t)chunk * 4;
        asm volatile("global_load_async_to_lds_b128 %0, %1, off"
                     :: "v"(laddr), "v"(ga) : "memory");
    }
    asm volatile("s_wait_asynccnt 0x0" ::: "memory");   // my waves' copies done
    __syncthreads();                                    // all waves' copies done

    if (nTile < NTILE) {
        const int n0 = nTile * 16;
        const float* bcol = Bmat + (size_t)n0 + l15;    // lane = N col
        const int aoff = l15 * ASTRIDE;                 // lane = M row in LDS

        v8f acc = (v8f)0.0f;
        #pragma unroll 8
        for (int k = 0; k < NFFT; k += 4) {
            // A 16x4 f32 layout: VGPR0 = K {0|2} per lane-half, VGPR1 = K {1|3}
            int kk = k + 2 * half;
            v2f av;
            av.x = smA[aoff + kk + 0];
            av.y = smA[aoff + kk + 1];
            // B 4x16 f32 dual layout: lane = N, VGPR0 = K {0|2}, VGPR1 = K {1|3}
            const float* bp = bcol + (size_t)kk * NPAD;
            v2f bv;
            bv.x = bp[0];
            bv.y = bp[NPAD];
            acc = __builtin_amdgcn_wmma_f32_16x16x4_f32(
                /*neg_a=*/false, av, /*neg_b=*/false, bv,
                /*c_mod=*/(short)0, acc, /*reuse_a=*/false, /*reuse_b=*/false);
        }

        // C/D layout: VGPR r, lanes 0-15 -> M=r ; lanes 16-31 -> M=r+8 ; N=lane&15
        const int col = n0 + l15;
        if (col < NOUT) {
            const int kbin = col >> 1;
            const int reim = col & 1;
            const int row_base = m0 + 8 * half;
            #pragma unroll
            for (int r = 0; r < 8; ++r) {
                int row = row_base + r;          // = b*800 + t
                int b   = row / TT;
                int t   = row - b * TT;
                out[(((size_t)b * NBINS + kbin) * TT + t) * 2 + reim] = acc[r];
            }
        }
    }
}

extern "C" void kernel_launch(void* const* d_in, const int* in_sizes, int n_in,
                              void* d_out, int out_size, void* d_ws, size_t ws_size,
                              hipStream_t stream) {
    (void)in_sizes; (void)n_in; (void)out_size; (void)ws_size;
    const float* f0   = (const float*)d_in[0];   // (4, 800)
    const float* amps = (const float*)d_in[1];   // (4, 800, 80)
    float* out = (float*)d_out;                  // (4, 513, 800) complex interleaved

    char* ws = (char*)d_ws;
    float*    audio  = (float*)(ws + OFF_AUDIO);
    float*    csum   = (float*)(ws + OFF_CSUM);
    float*    cbase  = (float*)(ws + OFF_CBASE);
    unsigned* maxbuf = (unsigned*)(ws + OFF_MAX);
    float*    Amat   = (float*)(ws + OFF_AMAT);
    float*    Bmat   = (float*)(ws + OFF_BMAT);

    // 1) per-chunk phase sums
    k_chunk_sums<<<dim3(NCH, BB), 256, 0, stream>>>(f0, csum);
    // 2) exclusive scan of chunk sums + init per-batch max
    k_scan<<<1, 32, 0, stream>>>(csum, cbase, maxbuf);
    // 3) harmonic synthesis with chunk-local scan + peak tracking
    k_synth<<<dim3(NCH, BB), 256, 0, stream>>>(f0, amps, cbase, audio, maxbuf);
    // 4) DFT coefficient matrix (independent)
    k_dft<<<NFFT, 256, 0, stream>>>(Bmat);
    // 5) normalize + reflect pad + window -> frame matrix
    k_frames<<<dim3(TT, BB), 256, 0, stream>>>(audio, maxbuf, Amat);
    // 6) STFT as f32 WMMA GEMM: LDS-staged A tile (async copy), 4 N-tiles/block
    k_gemm<<<dim3((NTILE + 3) / 4, MGEMM / 16), 128, 0, stream>>>(Amat, Bmat, out);
}